// GATv2Encoder_33861522162252
// MI455X (gfx1250) — compile-verified
//
#include <hip/hip_runtime.h>

typedef __attribute__((ext_vector_type(2))) float v2f;
typedef __attribute__((ext_vector_type(8))) float v8f;

#define IN_CH 128
#define HEADS 4
#define OUT_CH 32
#define KDIM 128            // HEADS*OUT_CH
#define NEG_SLOPE 0.2f

// ---------------------------------------------------------------------------
// Projection GEMM: out[N,128] = x[N,128] @ W[128,128]  (fp32 WMMA 16x16x4)
// grid.x = ceil(N/16) row tiles; 8 waves per block -> 8 column tiles (128 cols)
// ---------------------------------------------------------------------------
__global__ __launch_bounds__(256)
void proj_wmma(const float* __restrict__ x, const float* __restrict__ W,
               float* __restrict__ out, int N) {
  const int lane = threadIdx.x & 31;
  const int wave = threadIdx.x >> 5;
  const int m0   = blockIdx.x * 16;
  const int n0   = wave * 16;          // 8 waves * 16 = 128 output columns
  const int half = lane >> 4;          // 0 or 1
  const int l16  = lane & 15;

  // A layout (16x4 f32): lane holds row M = lane%16, K = 2*(lane/16) + vgpr
  int arow = m0 + l16; if (arow >= N) arow = N - 1;   // clamp (N%16==0 normally)
  const int koff = 2 * half;
  const float* xrow = x + (size_t)arow * IN_CH + koff;
  // B layout (4x16 f32): N = lane%16, K = 2*half + vgpr
  const float* wcol = W + (size_t)koff * KDIM + (n0 + l16);

  v8f acc = {};
#pragma unroll 8
  for (int k = 0; k < IN_CH; k += 4) {
    const float2 av = *(const float2*)(xrow + k);        // 8B-aligned b64 load
    v2f a; a.x = av.x; a.y = av.y;
    v2f b;
    b.x = wcol[(size_t)k * KDIM];
    b.y = wcol[(size_t)(k + 1) * KDIM];
    acc = __builtin_amdgcn_wmma_f32_16x16x4_f32(
        /*neg_a=*/false, a, /*neg_b=*/false, b,
        /*c_mod=*/(short)0, acc, /*reuse_a=*/false, /*reuse_b=*/false);
  }

  // C/D layout: row = m0 + vgpr + 8*(lane/16), col = n0 + lane%16
  const int crow = m0 + half * 8;
  float* po = out + (size_t)crow * KDIM + (n0 + l16);
  if (m0 + 16 <= N) {                  // uniform fast path: tile fully in-bounds
#pragma unroll
    for (int r = 0; r < 8; ++r) po[(size_t)r * KDIM] = acc[r];
  } else {                             // ragged tail (unused when N%16==0)
#pragma unroll
    for (int r = 0; r < 8; ++r)
      if (crow + r < N) po[(size_t)r * KDIM] = acc[r];
  }
}

// ---------------------------------------------------------------------------
// Init: out=0, m=-inf, denom=0
// ---------------------------------------------------------------------------
__global__ void init_kernel(float* __restrict__ out, float* __restrict__ mmax,
                            float* __restrict__ denom, int N) {
  int i = blockIdx.x * blockDim.x + threadIdx.x;
  if (i < N * KDIM) out[i] = 0.0f;
  if (i < N * HEADS) { mmax[i] = -__builtin_inff(); denom[i] = 0.0f; }
}

// float atomic max via monotone int encoding (valid for finite + -inf init)
__device__ __forceinline__ void atomicMaxF(float* addr, float v) {
  if (v >= 0.0f) atomicMax((int*)addr, __float_as_int(v));
  else           atomicMin((unsigned int*)addr, __float_as_uint(v));
}

// ---------------------------------------------------------------------------
// Edge pass 1: score[e,h] = sum_c att[h,c]*lrelu(xl[src,h,c]+xr[dst,h,c])
// one wave per edge; lane l covers flat channels l, l+32, l+64, l+96
// ---------------------------------------------------------------------------
__global__ __launch_bounds__(256)
void edge_score(const float* __restrict__ xl, const float* __restrict__ xr,
                const long long* __restrict__ ei, const float* __restrict__ att,
                float* __restrict__ mmax, float* __restrict__ scores,
                int E, int Etot) {
  const int wid = (int)(((size_t)blockIdx.x * blockDim.x + threadIdx.x) >> 5);
  if (wid >= Etot) return;
  const int lane = threadIdx.x & 31;

  long long src, dst;
  if (wid < E) { src = ei[wid]; dst = ei[(size_t)E + wid]; }
  else         { src = dst = wid - E; }

  const float* pl = xl + (size_t)src * KDIM;
  const float* pr = xr + (size_t)dst * KDIM;

  float s[HEADS];
#pragma unroll
  for (int j = 0; j < HEADS; ++j) {
    const int idx = j * 32 + lane;
    float v = pl[idx] + pr[idx];
    v = (v > 0.0f) ? v : NEG_SLOPE * v;
    s[j] = v * att[idx];
  }
#pragma unroll
  for (int j = 0; j < HEADS; ++j)
#pragma unroll
    for (int off = 16; off > 0; off >>= 1)
      s[j] += __shfl_xor(s[j], off, 32);

  if (lane == 0) {
#pragma unroll
    for (int j = 0; j < HEADS; ++j) {
      scores[(size_t)wid * HEADS + j] = s[j];
      atomicMaxF(&mmax[(size_t)dst * HEADS + j], s[j]);
    }
  }
}

// ---------------------------------------------------------------------------
// Edge pass 2: ex = exp(score - m[dst]); denom[dst] += ex   (thread per edge)
// ---------------------------------------------------------------------------
__global__ void edge_exp(const long long* __restrict__ ei,
                         const float* __restrict__ mmax,
                         float* __restrict__ scores, float* __restrict__ denom,
                         int E, int Etot) {
  const int e = blockIdx.x * blockDim.x + threadIdx.x;
  if (e >= Etot) return;
  const long long dst = (e < E) ? ei[(size_t)E + e] : (long long)(e - E);
#pragma unroll
  for (int j = 0; j < HEADS; ++j) {
    const float ex = __expf(scores[(size_t)e * HEADS + j] -
                            mmax[(size_t)dst * HEADS + j]);
    scores[(size_t)e * HEADS + j] = ex;
    atomicAdd(&denom[(size_t)dst * HEADS + j], ex);
  }
}

// ---------------------------------------------------------------------------
// Edge pass 3: out[dst] += (ex/denom[dst]) * xl[src]   (wave per edge)
// ---------------------------------------------------------------------------
__global__ __launch_bounds__(256)
void edge_aggr(const float* __restrict__ xl, const long long* __restrict__ ei,
               const float* __restrict__ scores, const float* __restrict__ denom,
               float* __restrict__ out, int E, int Etot) {
  const int wid = (int)(((size_t)blockIdx.x * blockDim.x + threadIdx.x) >> 5);
  if (wid >= Etot) return;
  const int lane = threadIdx.x & 31;

  long long src, dst;
  if (wid < E) { src = ei[wid]; dst = ei[(size_t)E + wid]; }
  else         { src = dst = wid - E; }

  float alpha[HEADS];
#pragma unroll
  for (int j = 0; j < HEADS; ++j)
    alpha[j] = scores[(size_t)wid * HEADS + j] /
               denom[(size_t)dst * HEADS + j];

  const float* pl = xl + (size_t)src * KDIM;
  float* po = out + (size_t)dst * KDIM;
#pragma unroll
  for (int j = 0; j < HEADS; ++j)
    atomicAdd(&po[j * 32 + lane], alpha[j] * pl[j * 32 + lane]);
}

// ---------------------------------------------------------------------------
// Bias add
// ---------------------------------------------------------------------------
__global__ void add_bias(float* __restrict__ out, const float* __restrict__ bias,
                         int N) {
  const int i = blockIdx.x * blockDim.x + threadIdx.x;
  if (i < N * KDIM) out[i] += bias[i & (KDIM - 1)];
}

// ---------------------------------------------------------------------------
extern "C" void kernel_launch(void* const* d_in, const int* in_sizes, int n_in,
                              void* d_out, int out_size, void* d_ws, size_t ws_size,
                              hipStream_t stream) {
  const float*     x    = (const float*)d_in[0];
  const long long* ei   = (const long long*)d_in[1];
  const float*     W_l  = (const float*)d_in[2];
  const float*     W_r  = (const float*)d_in[3];
  const float*     att  = (const float*)d_in[4];
  const float*     bias = (const float*)d_in[5];
  float*           out  = (float*)d_out;

  const int N    = in_sizes[0] / IN_CH;   // 50000
  const int E    = in_sizes[1] / 2;       // 800000
  const int Etot = E + N;                 // self loops appended

  // workspace layout (floats): xl | xr | scores | m | denom  (~67 MB)
  float* ws     = (float*)d_ws;
  float* xl     = ws;
  float* xr     = xl + (size_t)N * KDIM;
  float* scores = xr + (size_t)N * KDIM;
  float* mmax   = scores + (size_t)Etot * HEADS;
  float* denom  = mmax + (size_t)N * HEADS;

  const int mtiles = (N + 15) / 16;
  proj_wmma<<<mtiles, 256, 0, stream>>>(x, W_l, xl, N);
  proj_wmma<<<mtiles, 256, 0, stream>>>(x, W_r, xr, N);

  const int initTot = N * KDIM;
  init_kernel<<<(initTot + 255) / 256, 256, 0, stream>>>(out, mmax, denom, N);

  const int waveBlocks = (int)(((size_t)Etot * 32 + 255) / 256);
  edge_score<<<waveBlocks, 256, 0, stream>>>(xl, xr, ei, att, mmax, scores, E, Etot);
  edge_exp<<<(Etot + 255) / 256, 256, 0, stream>>>(ei, mmax, scores, denom, E, Etot);
  edge_aggr<<<waveBlocks, 256, 0, stream>>>(xl, ei, scores, denom, out, E, Etot);
  add_bias<<<(initTot + 255) / 256, 256, 0, stream>>>(out, bias, N);
}